// MoELayer_38534446579844
// MI455X (gfx1250) — compile-verified
//
#include <hip/hip_runtime.h>
#include <math.h>

typedef __attribute__((ext_vector_type(2))) float v2f;
typedef __attribute__((ext_vector_type(8))) float v8f;

static constexpr int kB = 4;     // batch
static constexpr int kS = 1024;  // seq
static constexpr int kH = 1024;  // hidden
static constexpr int kI = 4096;  // intermediate

__device__ __forceinline__ float gelu_exact(float v) {
    return 0.5f * v * (1.0f + erff(v * 0.70710678118654752f));
}

__device__ __forceinline__ v8f wmma_f32(v2f a, v2f b, v8f c) {
    return __builtin_amdgcn_wmma_f32_16x16x4_f32(
        false, a, false, b, (short)0, c, false, false);
}

// low 32 bits of a flat shared-aperture address == LDS byte offset (ISA 10.2)
__device__ __forceinline__ uint32_t lds_off(const void* p) {
    return (uint32_t)(uintptr_t)p;
}

// Async copy engine: global -> LDS, GVS addressing, tracked by ASYNCcnt.
__device__ __forceinline__ void async_ld_b128(uint32_t lds, const void* base, uint32_t off) {
    asm volatile("global_load_async_to_lds_b128 %0, %1, %2"
                 :: "v"(lds), "v"(off), "s"(base) : "memory");
}
__device__ __forceinline__ void async_ld_b32(uint32_t lds, const void* base, uint32_t off) {
    asm volatile("global_load_async_to_lds_b32 %0, %1, %2"
                 :: "v"(lds), "v"(off), "s"(base) : "memory");
}

// ---------------------------------------------------------------------------
// Kernel 1: h = gelu(x @ Wi[e] + bi[e])
// grid=(kI/128, kS/64, kB), block=256. Double-buffered async K panels (K=16).
// ---------------------------------------------------------------------------
static constexpr int AST1 = 20;   // A stride: rows 16B-aligned, ord(20 mod 64)=16 -> conflict-free
static constexpr int BST1 = 136;  // B stride: 544B rows (16B aligned), phase-shifts half-waves

__global__ __launch_bounds__(256)
void moe_gemm1_gelu(const float* __restrict__ x, const int* __restrict__ eidx,
                    const float* __restrict__ Wi, const float* __restrict__ bi,
                    float* __restrict__ h) {
    __shared__ float As[2][64 * AST1];
    __shared__ float Bs[2][16 * BST1];

    const int tid  = threadIdx.x;
    const int lane = tid & 31;
    const int w    = tid >> 5;
    const int wm   = w >> 1;          // 0..3
    const int wn   = w & 1;           // 0..1
    const int b    = blockIdx.z;
    const int e    = eidx[b];
    const int n0   = blockIdx.x * 128;
    const int m0   = blockIdx.y * 64;

    const float* xb  = x  + (size_t)b * kS * kH;
    const float* Wie = Wi + (size_t)e * kH * kI;

    const int mrow = lane & 15;
    const int khi  = (lane >> 4) << 1;

    // A: 64x16 as 256 float4 (1 per thread); B: 16x128 as 512 float4 (2 per thread)
    auto issue_panel = [&](int buf, int k0) {
        {
            int r = tid >> 2, s4 = tid & 3;
            async_ld_b128(lds_off(&As[buf][r * AST1 + s4 * 4]), xb,
                          (uint32_t)((((m0 + r) * kH) + k0 + s4 * 4) * 4));
        }
        #pragma unroll
        for (int i = 0; i < 2; ++i) {
            int idx = tid + i * 256;
            int r = idx >> 5, seg = idx & 31;
            async_ld_b128(lds_off(&Bs[buf][r * BST1 + seg * 4]), Wie,
                          (uint32_t)((((k0 + r) * kI) + n0 + seg * 4) * 4));
        }
    };

    v8f acc[4] = {};

    issue_panel(0, 0);
    for (int k0 = 0; k0 < kH; k0 += 16) {
        const int p = (k0 >> 4) & 1;
        const bool more = (k0 + 16) < kH;
        if (more) {
            issue_panel(p ^ 1, k0 + 16);
            asm volatile("s_wait_asynccnt 0x3" ::: "memory");  // newest 3 pending -> panel p done
        } else {
            asm volatile("s_wait_asynccnt 0x0" ::: "memory");
        }
        __syncthreads();

        #pragma unroll
        for (int kk = 0; kk < 16; kk += 4) {
            v2f a;
            a.x = As[p][(wm * 16 + mrow) * AST1 + kk + khi];
            a.y = As[p][(wm * 16 + mrow) * AST1 + kk + khi + 1];
            #pragma unroll
            for (int t = 0; t < 4; ++t) {
                int col = wn * 64 + t * 16 + mrow;
                v2f bfr;
                bfr.x = Bs[p][(kk + khi) * BST1 + col];
                bfr.y = Bs[p][(kk + khi + 1) * BST1 + col];
                acc[t] = wmma_f32(a, bfr, acc[t]);
            }
        }
        __syncthreads();   // all waves done with panel p before it is overwritten
    }

    const int rbase = m0 + wm * 16 + ((lane >> 4) << 3);
    #pragma unroll
    for (int t = 0; t < 4; ++t) {
        int col = n0 + wn * 64 + t * 16 + mrow;
        float bias = bi[(size_t)e * kI + col];
        #pragma unroll
        for (int v = 0; v < 8; ++v) {
            float val = acc[t][v] + bias;
            h[((size_t)b * kS + rbase + v) * kI + col] = gelu_exact(val);
        }
    }
}

// ---------------------------------------------------------------------------
// Kernel 2: o = h @ Wo[e] + bo[e] + x ; LayerNorm(H) ; *gamma + beta
// grid=(kS/16, kB), block=256. 16 rows x full 1024 cols; double-buffered async.
// ---------------------------------------------------------------------------
static constexpr int AST2 = 17;
static constexpr int BST2 = 1032;  // 4128B rows: 16B aligned, half-waves bank-disjoint

__global__ __launch_bounds__(256)
void moe_gemm2_ln(const float* __restrict__ x, const int* __restrict__ eidx,
                  const float* __restrict__ Wo, const float* __restrict__ bo,
                  const float* __restrict__ gamma, const float* __restrict__ beta,
                  const float* __restrict__ h, float* __restrict__ out) {
    __shared__ float As[2][16 * AST2];
    __shared__ float Bs[2][16 * BST2];   // buffer 0 reused as r[16][1024] in epilogue
    __shared__ float redS[256], redQ[256];
    __shared__ float muA[16], rsA[16];

    const int tid  = threadIdx.x;
    const int lane = tid & 31;
    const int w    = tid >> 5;           // N base = w*128
    const int b    = blockIdx.y;
    const int e    = eidx[b];
    const int s0   = blockIdx.x * 16;

    const float* hb  = h  + ((size_t)b * kS + s0) * kI;
    const float* Woe = Wo + (size_t)e * kI * kH;

    const int mrow = lane & 15;
    const int khi  = (lane >> 4) << 1;

    // A: 16x16 = 256 floats (1 b32 per thread); B: 16x1024 = 4096 float4 (16 per thread)
    auto issue_panel = [&](int buf, int k0) {
        {
            int r = tid >> 4, c = tid & 15;
            async_ld_b32(lds_off(&As[buf][r * AST2 + c]), hb,
                         (uint32_t)((r * kI + k0 + c) * 4));
        }
        #pragma unroll
        for (int i = 0; i < 16; ++i) {
            int idx = tid + i * 256;
            int r = idx >> 8, seg = idx & 255;
            async_ld_b128(lds_off(&Bs[buf][r * BST2 + seg * 4]), Woe,
                          (uint32_t)((((k0 + r) * kH) + seg * 4) * 4));
        }
    };

    v8f acc[8] = {};

    issue_panel(0, 0);
    for (int k0 = 0; k0 < kI; k0 += 16) {
        const int p = (k0 >> 4) & 1;
        const bool more = (k0 + 16) < kI;
        if (more) {
            issue_panel(p ^ 1, k0 + 16);
            asm volatile("s_wait_asynccnt 0x11" ::: "memory");  // 17 newer pending -> panel p done
        } else {
            asm volatile("s_wait_asynccnt 0x0" ::: "memory");
        }
        __syncthreads();

        #pragma unroll
        for (int kk = 0; kk < 16; kk += 4) {
            v2f a;
            a.x = As[p][mrow * AST2 + kk + khi];
            a.y = As[p][mrow * AST2 + kk + khi + 1];
            #pragma unroll
            for (int t = 0; t < 8; ++t) {
                int col = w * 128 + t * 16 + mrow;
                v2f bfr;
                bfr.x = Bs[p][(kk + khi) * BST2 + col];
                bfr.y = Bs[p][(kk + khi + 1) * BST2 + col];
                acc[t] = wmma_f32(a, bfr, acc[t]);
            }
        }
        __syncthreads();
    }

    // dump accumulators into LDS r[16][1024] (reuse Bs[0])
    float* R = &Bs[0][0];
    {
        int rtop = (lane >> 4) << 3;
        #pragma unroll
        for (int t = 0; t < 8; ++t) {
            int col = w * 128 + t * 16 + mrow;
            #pragma unroll
            for (int v = 0; v < 8; ++v)
                R[(rtop + v) * BST2 + col] = acc[t][v];
        }
    }
    __syncthreads();

    // fused bias + residual + LayerNorm; 16 threads per row, coalesced strided cols
    {
        int row = tid >> 4, tc = tid & 15;
        const float* xrow = x + ((size_t)b * kS + s0 + row) * kH;
        float s = 0.f, q = 0.f;
        for (int j = 0; j < 64; ++j) {
            int col = tc + j * 16;
            float v = R[row * BST2 + col] + bo[(size_t)e * kH + col] + xrow[col];
            R[row * BST2 + col] = v;
            s += v; q += v * v;
        }
        redS[tid] = s; redQ[tid] = q;
        __syncthreads();
        if (tc == 0) {
            float ss = 0.f, qq = 0.f;
            #pragma unroll
            for (int i = 0; i < 16; ++i) { ss += redS[row * 16 + i]; qq += redQ[row * 16 + i]; }
            float mu  = ss * (1.0f / 1024.0f);
            float var = qq * (1.0f / 1024.0f) - mu * mu;
            muA[row] = mu;
            rsA[row] = rsqrtf(var + 1e-12f);
        }
        __syncthreads();
        float mu = muA[row], rstd = rsA[row];
        float* orow = out + ((size_t)b * kS + s0 + row) * kH;
        const float* grow = gamma + (size_t)e * kH;
        const float* brow = beta  + (size_t)e * kH;
        for (int j = 0; j < 64; ++j) {
            int col = tc + j * 16;
            float v = R[row * BST2 + col];
            orow[col] = (v - mu) * rstd * grow[col] + brow[col];
        }
    }
}

extern "C" void kernel_launch(void* const* d_in, const int* in_sizes, int n_in,
                              void* d_out, int out_size, void* d_ws, size_t ws_size,
                              hipStream_t stream) {
    const float* x     = (const float*)d_in[0];
    const int*   eidx  = (const int*)  d_in[1];
    const float* Wi    = (const float*)d_in[2];
    const float* bi    = (const float*)d_in[3];
    const float* Wo    = (const float*)d_in[4];
    const float* bo    = (const float*)d_in[5];
    const float* gamma = (const float*)d_in[6];
    const float* beta  = (const float*)d_in[7];
    float* out = (float*)d_out;
    float* h   = (float*)d_ws;   // [kB, kS, kI] fp32 = 64 MB scratch

    dim3 g1(kI / 128, kS / 64, kB);
    moe_gemm1_gelu<<<g1, 256, 0, stream>>>(x, eidx, Wi, bi, h);

    dim3 g2(kS / 16, kB);
    moe_gemm2_ln<<<g2, 256, 0, stream>>>(x, eidx, Wo, bo, gamma, beta, h, out);
}